// SparseVariationalPooler_35330400977496
// MI455X (gfx1250) — compile-verified
//
#include <hip/hip_runtime.h>

#define BOOSTP 1e-8f
#define HB 256
#define HW (HB / 32)

struct Ctl {
  unsigned int maxKey;
  unsigned int prefixA, needA, tieTicketA;
  unsigned int prefixB, needB, tieTicketB;
  unsigned int activeCount;
};

// ---- monotone float <-> uint key (descending select on uint order) ----
__device__ __forceinline__ unsigned int f2ord(float f) {
  unsigned int u = __float_as_uint(f);
  return (u & 0x80000000u) ? ~u : (u | 0x80000000u);
}
__device__ __forceinline__ float ord2f(unsigned int k) {
  unsigned int u = (k & 0x80000000u) ? (k ^ 0x80000000u) : ~k;
  return __uint_as_float(u);
}

// ---- CDNA5 async global->LDS (ASYNCcnt) helpers ----
typedef int v4i_t __attribute__((vector_size(16)));

__device__ __forceinline__ void async_copy_b128(const float* g, float* lds) {
#if defined(__HIP_DEVICE_COMPILE__)
#if __has_builtin(__builtin_amdgcn_global_load_async_to_lds_b128)
  __builtin_amdgcn_global_load_async_to_lds_b128(
      (__attribute__((address_space(1))) v4i_t*)g,
      (__attribute__((address_space(3))) v4i_t*)lds,
      /*imm offset=*/0, /*cpol=*/0);
#else
  unsigned int l = (unsigned int)(unsigned long long)lds;
  asm volatile("global_load_async_to_lds_b128 %0, %1, off" ::"v"(l), "v"(g)
               : "memory");
#endif
#endif
}
__device__ __forceinline__ void wait_async0() {
#if defined(__HIP_DEVICE_COMPILE__)
#if __has_builtin(__builtin_amdgcn_s_wait_asynccnt)
  __builtin_amdgcn_s_wait_asynccnt(0);
#else
  asm volatile("s_wait_asynccnt 0x0" ::: "memory");
#endif
#endif
}
__device__ __forceinline__ void wait_async1() {
#if defined(__HIP_DEVICE_COMPILE__)
#if __has_builtin(__builtin_amdgcn_s_wait_asynccnt)
  __builtin_amdgcn_s_wait_asynccnt(1);
#else
  asm volatile("s_wait_asynccnt 0x1" ::: "memory");
#endif
#endif
}

// ---------------- init: zero control + 6x2048 histograms ----------------
__global__ void k_init(Ctl* ctl, unsigned int* hist, unsigned int maxActive) {
  for (unsigned int i = threadIdx.x; i < 6u * 2048u; i += blockDim.x) hist[i] = 0u;
  if (threadIdx.x < sizeof(Ctl) / 4) ((unsigned int*)ctl)[threadIdx.x] = 0u;
  __syncthreads();
  if (threadIdx.x == 0) ctl->needA = maxActive;
}

// ---------------- global max of x (wave32 shuffle + atomic) -------------
__global__ void __launch_bounds__(256) k_max(const float* __restrict__ x, Ctl* ctl,
                                             unsigned int n) {
  unsigned int tid = blockIdx.x * blockDim.x + threadIdx.x;
  unsigned int stride = gridDim.x * blockDim.x;
  unsigned int n4 = n >> 2;
  const float4* x4 = (const float4*)x;
  float m = -3.402823466e38f;
  for (unsigned int i = tid; i < n4; i += stride) {
    __builtin_prefetch(x4 + i + 2u * stride, 0, 0);  // global_prefetch_b8
    float4 v = x4[i];
    m = fmaxf(m, fmaxf(fmaxf(v.x, v.y), fmaxf(v.z, v.w)));
  }
  for (unsigned int i = (n4 << 2) + tid; i < n; i += stride) m = fmaxf(m, x[i]);
  for (int o = 16; o > 0; o >>= 1) m = fmaxf(m, __shfl_xor(m, o, 32));
  __shared__ float wred[8];
  unsigned int lane = threadIdx.x & 31u, wave = threadIdx.x >> 5u;
  if (lane == 0) wred[wave] = m;
  __syncthreads();
  if (threadIdx.x == 0) {
    float bm = wred[0];
    for (int w = 1; w < 8; ++w) bm = fmaxf(bm, wred[w]);
    atomicMax(&ctl->maxKey, f2ord(bm));
  }
}

// ---- boost+relu, store inhibited & boost, fused pass-0 histograms ------
__global__ void __launch_bounds__(256) k_prep(const float* __restrict__ x,
                                              const float* __restrict__ bf,
                                              float* __restrict__ inh,
                                              float* __restrict__ boost,
                                              unsigned int* __restrict__ histA0,
                                              unsigned int* __restrict__ histB0,
                                              const Ctl* __restrict__ ctl,
                                              unsigned int n) {
  __shared__ unsigned int lhA[2048];
  __shared__ unsigned int lhB[2048];
  for (unsigned int i = threadIdx.x; i < 2048; i += blockDim.x) { lhA[i] = 0u; lhB[i] = 0u; }
  __syncthreads();
  const float maxv = ord2f(ctl->maxKey);
  const float inv = 1.0f / maxv;
  unsigned int tid = blockIdx.x * blockDim.x + threadIdx.x;
  unsigned int stride = gridDim.x * blockDim.x;
  unsigned int n4 = n >> 2;
  const float4* x4 = (const float4*)x;
  const float4* b4 = (const float4*)bf;
  float4* i4 = (float4*)inh;
  float4* o4 = (float4*)boost;
  for (unsigned int i = tid; i < n4; i += stride) {
    float4 xv = x4[i], bv = b4[i], so, bo;
#define PREP1(c)                                                 \
  {                                                              \
    bo.c = bv.c + (1.0f - xv.c * inv) * BOOSTP;                  \
    so.c = fmaxf(xv.c, 0.0f) + bo.c;                             \
    atomicAdd(&lhA[f2ord(so.c) >> 21], 1u);                      \
    atomicAdd(&lhB[f2ord(bo.c) >> 21], 1u);                      \
  }
    PREP1(x) PREP1(y) PREP1(z) PREP1(w)
#undef PREP1
    i4[i] = so;
    o4[i] = bo;
  }
  for (unsigned int i = (n4 << 2) + tid; i < n; i += stride) {
    float bo = bf[i] + (1.0f - x[i] * inv) * BOOSTP;
    float so = fmaxf(x[i], 0.0f) + bo;
    inh[i] = so; boost[i] = bo;
    atomicAdd(&lhA[f2ord(so) >> 21], 1u);
    atomicAdd(&lhB[f2ord(bo) >> 21], 1u);
  }
  __syncthreads();
  for (unsigned int i = threadIdx.x; i < 2048; i += blockDim.x) {
    if (lhA[i]) atomicAdd(&histA0[i], lhA[i]);
    if (lhB[i]) atomicAdd(&histB0[i], lhB[i]);
  }
}

// ---- sparse self-affector: inh[affectee] += boosted(affector)*val ------
// Recomputes boosted(affector) from x/bf (pre-update semantics) and
// incrementally repairs the pass-0 histogram using the atomic pre-op value.
__global__ void k_scatter(const float* __restrict__ x, const float* __restrict__ bf,
                          const float* __restrict__ vals, const int* __restrict__ aff,
                          const int* __restrict__ afe, float* __restrict__ inh,
                          unsigned int* __restrict__ histA0,
                          const Ctl* __restrict__ ctl, unsigned int nnz) {
  unsigned int i = blockIdx.x * blockDim.x + threadIdx.x;
  if (i >= nnz) return;
  const float maxv = ord2f(ctl->maxKey);
  int a = aff[i], e = afe[i];
  float xa = x[a];
  float bo = bf[a] + (1.0f - xa / maxv) * BOOSTP;
  float c = (fmaxf(xa, 0.0f) + bo) * vals[i];
  float old = atomicAdd(&inh[e], c);
  unsigned int b0 = f2ord(old) >> 21;
  unsigned int b1 = f2ord(old + c) >> 21;
  if (b0 != b1) {
    atomicSub(&histA0[b0], 1u);
    atomicAdd(&histA0[b1], 1u);
  }
}

// ---- conditional histogram pass (PASS=1: mid 11 bits, PASS=2: low 10) --
// Streams the array through LDS with double-buffered async loads (ASYNCcnt).
template <int PASS>
__global__ void __launch_bounds__(HB) k_hist(const float* __restrict__ src,
                                             unsigned int* __restrict__ hist,
                                             const Ctl* __restrict__ ctl, int selB,
                                             unsigned int n) {
  __shared__ unsigned int lh[2048];
  __shared__ alignas(16) float stage[2][HW][128];
  for (unsigned int i = threadIdx.x; i < 2048; i += HB) lh[i] = 0u;
  __syncthreads();
  const unsigned int prefix = selB ? ctl->prefixB : ctl->prefixA;
  const unsigned int lane = threadIdx.x & 31u;
  const unsigned int wave = threadIdx.x >> 5u;
  const unsigned int tiles = n >> 7;  // 128 floats per wave-tile
  const unsigned int gwave = blockIdx.x * HW + wave;
  const unsigned int wstride = gridDim.x * HW;

  auto acc1 = [&](float f) {
    unsigned int key = f2ord(f);
    if (PASS == 1) {
      if ((key >> 21) == (prefix >> 21)) atomicAdd(&lh[(key >> 10) & 0x7FFu], 1u);
    } else {
      if ((key >> 10) == (prefix >> 10)) atomicAdd(&lh[key & 0x3FFu], 1u);
    }
  };

  if (gwave < tiles) {
    async_copy_b128(src + (size_t)gwave * 128 + lane * 4, &stage[0][wave][lane * 4]);
    int buf = 0;
    for (unsigned int t = gwave; t < tiles; t += wstride) {
      unsigned int tn = t + wstride;
      if (tn < tiles) {
        async_copy_b128(src + (size_t)tn * 128 + lane * 4,
                        &stage[buf ^ 1][wave][lane * 4]);
        wait_async1();  // oldest (current tile) complete
      } else {
        wait_async0();
      }
      float4 v = *(const float4*)&stage[buf][wave][lane * 4];
      acc1(v.x); acc1(v.y); acc1(v.z); acc1(v.w);
      buf ^= 1;
    }
  }
  for (unsigned int i = (tiles << 7) + blockIdx.x * HB + threadIdx.x; i < n;
       i += gridDim.x * HB)
    acc1(src[i]);
  __syncthreads();
  for (unsigned int i = threadIdx.x; i < 2048; i += HB) {
    unsigned int c = lh[i];
    if (c) atomicAdd(&hist[i], c);
  }
}

// ---- parallel descending-rank scan: refine prefix, remaining need ------
// One 1024-thread block: LDS suffix-sum (Hillis-Steele) over nbins, then the
// unique crossing bin b (suffix[b] >= need > suffix[b+1]) updates Ctl.
__global__ void __launch_bounds__(1024) k_scan(const unsigned int* __restrict__ hist,
                                               Ctl* ctl, int nbins, int shift,
                                               int selB) {
  __shared__ unsigned int s[2048];
  const unsigned int need = selB ? ctl->needB : ctl->needA;
  const unsigned int prefix = selB ? ctl->prefixB : ctl->prefixA;
  if (need == 0u) {  // nothing to select: sentinel threshold selects nothing
    if (threadIdx.x == 0) {
      if (selB) { ctl->prefixB = 0xFFFFFFFFu; ctl->needB = 0u; }
      else      { ctl->prefixA = 0xFFFFFFFFu; ctl->needA = 0u; }
    }
    return;
  }
  const int b0 = threadIdx.x, b1 = threadIdx.x + 1024;
  for (int b = b0; b < nbins; b += 1024) s[b] = hist[b];
  __syncthreads();
  for (int d = 1; d < nbins; d <<= 1) {
    unsigned int v0 = (b0 < nbins && b0 + d < nbins) ? s[b0 + d] : 0u;
    unsigned int v1 = (b1 < nbins && b1 + d < nbins) ? s[b1 + d] : 0u;
    __syncthreads();
    if (b0 < nbins) s[b0] += v0;
    if (b1 < nbins) s[b1] += v1;
    __syncthreads();
  }
  // s[b] = count of elements with bin >= b (conditioned on current prefix)
  for (int b = b0; b < nbins; b += 1024) {
    unsigned int suf = s[b];
    unsigned int sufNext = (b + 1 < nbins) ? s[b + 1] : 0u;
    bool cross = (suf >= need) && (sufNext < need);
    if (b == 0 && s[0] < need) { cross = true; sufNext = 0u; }  // degenerate
    if (cross) {
      unsigned int np = prefix | ((unsigned int)b << shift);
      unsigned int nn = need - sufNext;  // remaining rank inside chosen bin
      if (selB) { ctl->prefixB = np; ctl->needB = nn; }
      else      { ctl->prefixA = np; ctl->needA = nn; }
    }
  }
}

// ---- binarize top-K positives, count actives ---------------------------
__global__ void __launch_bounds__(256) k_applyA(const float* __restrict__ inh,
                                                float* __restrict__ out, Ctl* ctl,
                                                unsigned int n) {
  const unsigned int T = ctl->prefixA;   // exact 32-bit key threshold
  const unsigned int tie = ctl->needA;   // #ties at T to accept
  unsigned int cnt = 0;
  unsigned int tid = blockIdx.x * blockDim.x + threadIdx.x;
  unsigned int stride = gridDim.x * blockDim.x;
  unsigned int n4 = n >> 2;
  const float4* in4 = (const float4*)inh;
  float4* out4 = (float4*)out;
  for (unsigned int i = tid; i < n4; i += stride) {
    float4 v = in4[i], r;
#define SEL1(c)                                                        \
  {                                                                    \
    unsigned int k = f2ord(v.c);                                       \
    int s = (k > T);                                                   \
    if (!s && k == T) s = (atomicAdd(&ctl->tieTicketA, 1u) < tie);     \
    r.c = (s && (v.c > 0.0f)) ? 1.0f : 0.0f;                           \
    cnt += (r.c != 0.0f);                                              \
  }
    SEL1(x) SEL1(y) SEL1(z) SEL1(w)
#undef SEL1
    out4[i] = r;
  }
  for (unsigned int i = (n4 << 2) + tid; i < n; i += stride) {
    float v = inh[i];
    unsigned int k = f2ord(v);
    int s = (k > T);
    if (!s && k == T) s = (atomicAdd(&ctl->tieTicketA, 1u) < tie);
    float r = (s && (v > 0.0f)) ? 1.0f : 0.0f;
    out[i] = r;
    cnt += (r != 0.0f);
  }
  for (int o = 16; o > 0; o >>= 1) cnt += __shfl_xor(cnt, o, 32);
  if ((threadIdx.x & 31u) == 0) atomicAdd(&ctl->activeCount, cnt);
}

// ---- to_activate = max(MIN_ACTIVE - actually_active, 0) ----------------
__global__ void k_needB(Ctl* ctl, unsigned int minActive) {
  unsigned int act = ctl->activeCount;
  ctl->prefixB = 0u;
  ctl->needB = (act < minActive) ? (minActive - act) : 0u;
}

// ---- backfill: set out=1 for top-to_activate of boost ------------------
__global__ void __launch_bounds__(256) k_applyB(const float* __restrict__ boost,
                                                float* __restrict__ out, Ctl* ctl,
                                                unsigned int n) {
  const unsigned int T = ctl->prefixB;
  const unsigned int tie = ctl->needB;
  unsigned int tid = blockIdx.x * blockDim.x + threadIdx.x;
  unsigned int stride = gridDim.x * blockDim.x;
  unsigned int n4 = n >> 2;
  const float4* b4 = (const float4*)boost;
  for (unsigned int i = tid; i < n4; i += stride) {
    float4 v = b4[i];
#define SEL1(c, off)                                                   \
  {                                                                    \
    unsigned int k = f2ord(v.c);                                       \
    int s = (k > T);                                                   \
    if (!s && k == T) s = (atomicAdd(&ctl->tieTicketB, 1u) < tie);     \
    if (s) out[i * 4u + off] = 1.0f;                                   \
  }
    SEL1(x, 0u) SEL1(y, 1u) SEL1(z, 2u) SEL1(w, 3u)
#undef SEL1
  }
  for (unsigned int i = (n4 << 2) + tid; i < n; i += stride) {
    unsigned int k = f2ord(boost[i]);
    int s = (k > T);
    if (!s && k == T) s = (atomicAdd(&ctl->tieTicketB, 1u) < tie);
    if (s) out[i] = 1.0f;
  }
}

extern "C" void kernel_launch(void* const* d_in, const int* in_sizes, int n_in,
                              void* d_out, int out_size, void* d_ws, size_t ws_size,
                              hipStream_t stream) {
  const float* x    = (const float*)d_in[0];
  const float* bf   = (const float*)d_in[1];
  const float* vals = (const float*)d_in[2];
  const int* aff    = (const int*)d_in[3];
  const int* afe    = (const int*)d_in[4];
  float* out        = (float*)d_out;
  const unsigned int E   = (unsigned int)in_sizes[0];
  const unsigned int NNZ = (unsigned int)in_sizes[2];
  const unsigned int MAXA = (E + 49u) / 50u;  // ceil(0.02*E)  = 167773
  const unsigned int MINA = E / 500u;         // floor(0.002*E) = 16777

  char* ws = (char*)d_ws;
  Ctl* ctl = (Ctl*)ws;                                     // [0,32)
  unsigned int* hist = (unsigned int*)(ws + 256);          // 6 x 2048 u32
  float* inh   = (float*)(ws + 65536);                     // E floats
  float* boost = (float*)(ws + 65536 + (size_t)E * 4u);    // E floats

  k_init<<<1, 256, 0, stream>>>(ctl, hist, MAXA);
  k_max<<<1024, 256, 0, stream>>>(x, ctl, E);
  k_prep<<<2048, 256, 0, stream>>>(x, bf, inh, boost, hist + 0, hist + 3 * 2048,
                                   ctl, E);
  k_scatter<<<(NNZ + 255u) / 256u, 256, 0, stream>>>(x, bf, vals, aff, afe, inh,
                                                     hist + 0, ctl, NNZ);
  // ---- radix select A: K-th largest of inhibited (pass0 fused above) ----
  k_scan<<<1, 1024, 0, stream>>>(hist + 0, ctl, 2048, 21, 0);
  k_hist<1><<<1024, HB, 0, stream>>>(inh, hist + 1 * 2048, ctl, 0, E);
  k_scan<<<1, 1024, 0, stream>>>(hist + 1 * 2048, ctl, 2048, 10, 0);
  k_hist<2><<<1024, HB, 0, stream>>>(inh, hist + 2 * 2048, ctl, 0, E);
  k_scan<<<1, 1024, 0, stream>>>(hist + 2 * 2048, ctl, 1024, 0, 0);
  k_applyA<<<2048, 256, 0, stream>>>(inh, out, ctl, E);
  // ---- radix select B: top to_activate of boost (pass0 fused in prep) ---
  k_needB<<<1, 1, 0, stream>>>(ctl, MINA);
  k_scan<<<1, 1024, 0, stream>>>(hist + 3 * 2048, ctl, 2048, 21, 1);
  k_hist<1><<<1024, HB, 0, stream>>>(boost, hist + 4 * 2048, ctl, 1, E);
  k_scan<<<1, 1024, 0, stream>>>(hist + 4 * 2048, ctl, 2048, 10, 1);
  k_hist<2><<<1024, HB, 0, stream>>>(boost, hist + 5 * 2048, ctl, 1, E);
  k_scan<<<1, 1024, 0, stream>>>(hist + 5 * 2048, ctl, 1024, 0, 1);
  k_applyB<<<2048, 256, 0, stream>>>(boost, out, ctl, E);
}